// QuantumLinear_12987981103238
// MI455X (gfx1250) — compile-verified
//
#include <hip/hip_runtime.h>
#include <hip/hip_bf16.h>

// B=4096, IN_F=1024, NQ=4, L=2, C=256
#define NQB   4096
#define NCIR  256
#define NLAY  2
#define NQ    4

typedef float v2f __attribute__((ext_vector_type(2)));
typedef float v8f __attribute__((ext_vector_type(8)));

// ---------------------------------------------------------------------------
// Phase 1: per (circuit c, basis column k) simulate the entangler unitary on
// |k>, apply column phase (-i)^popcount(k), store P=Re(W), Q=Im(W) row-major:
//   PQ[c*512 +       m*16 + k] = P[m][k]
//   PQ[c*512 + 256 + m*16 + k] = Q[m][k]
// ---------------------------------------------------------------------------
__global__ void build_pq_kernel(const float* __restrict__ weights,
                                float* __restrict__ PQ) {
    int tid = blockIdx.x * blockDim.x + threadIdx.x;   // 0 .. 4095
    if (tid >= NCIR * 16) return;
    int c = tid >> 4;
    int k = tid & 15;

    float sr[16], si[16];
#pragma unroll
    for (int m = 0; m < 16; ++m) { sr[m] = 0.0f; si[m] = 0.0f; }
    sr[k] = 1.0f;

#pragma unroll
    for (int l = 0; l < NLAY; ++l) {
        // RX(w[l][q]) on each wire q (qubit 0 = MSB of state index)
#pragma unroll
        for (int q = 0; q < NQ; ++q) {
            float th = weights[(c * NLAY + l) * NQ + q] * 0.5f;
            float ss, cc;
            __sincosf(th, &ss, &cc);
            int mask = 1 << (3 - q);
#pragma unroll
            for (int m0 = 0; m0 < 16; ++m0) {
                if (m0 & mask) continue;
                int m1 = m0 | mask;
                float ar = sr[m0], ai = si[m0];
                float br = sr[m1], bi = si[m1];
                // |0>: c*a - i*s*b ; |1>: -i*s*a + c*b
                sr[m0] = cc * ar + ss * bi;
                si[m0] = cc * ai - ss * br;
                sr[m1] = cc * br + ss * ai;
                si[m1] = cc * bi - ss * ar;
            }
        }
        // ring of CNOTs: ctrl q -> tgt (q+1)%4
#pragma unroll
        for (int q = 0; q < NQ; ++q) {
            int cmask = 1 << (3 - q);
            int tmask = 1 << (3 - ((q + 1) & 3));
#pragma unroll
            for (int m = 0; m < 16; ++m) {
                if ((m & cmask) && !(m & tmask)) {
                    int m2 = m | tmask;
                    float tr = sr[m], ti = si[m];
                    sr[m] = sr[m2]; si[m] = si[m2];
                    sr[m2] = tr;    si[m2] = ti;
                }
            }
        }
    }

    // column phase (-i)^popcount(k)
    int pk = __popc(k) & 3;
    float pr = (pk == 0) ? 1.0f : ((pk == 2) ? -1.0f : 0.0f);
    float pi = (pk == 1) ? -1.0f : ((pk == 3) ? 1.0f : 0.0f);

    float* Pb = PQ + c * 512;
#pragma unroll
    for (int m = 0; m < 16; ++m) {
        float wr = sr[m] * pr - si[m] * pi;
        float wi = sr[m] * pi + si[m] * pr;
        Pb[m * 16 + k]       = wr;   // P
        Pb[256 + m * 16 + k] = wi;   // Q
    }
}

// ---------------------------------------------------------------------------
// Phase 2: one wave32 per (circuit c, tile of 16 batch rows).
//   R (16k x 16b) tile in LDS; u = P*R, v = Q*R via chained
//   V_WMMA_F32_16X16X4_F32; out[b, 4c+q] = sum_m z_q(m) (u_m^2 + v_m^2).
// ---------------------------------------------------------------------------
__global__ void __launch_bounds__(256)
quantum_expval_kernel(const float* __restrict__ x,
                      const float* __restrict__ PQ,
                      float* __restrict__ out) {
    __shared__ float s_r[8][256];          // per-wave 16x16 R tile, [k][b]

    int wave = threadIdx.x >> 5;
    int lane = threadIdx.x & 31;
    int t = blockIdx.x * 8 + wave;         // tile id: 0 .. 65535
    int c     = t & (NCIR - 1);
    int btile = t >> 8;
    int b0    = btile * 16;

    float* R = s_r[wave];

    // ---- build R tile: lane L (<16) handles batch row b0+L -----------------
    if (lane < 16) {
        int b = b0 + lane;
        const float4 xv = *(const float4*)(x + (size_t)b * 1024 + c * 4);
        float s0, c0, s1, c1, s2, c2, s3, c3;
        __sincosf(xv.x * 0.5f, &s0, &c0);
        __sincosf(xv.y * 0.5f, &s1, &c1);
        __sincosf(xv.z * 0.5f, &s2, &c2);
        __sincosf(xv.w * 0.5f, &s3, &c3);
        float t0[2] = {c0, s0}, t1[2] = {c1, s1};
        float t2[2] = {c2, s2}, t3[2] = {c3, s3};
#pragma unroll
        for (int k = 0; k < 16; ++k) {
            float r = t0[(k >> 3) & 1] * t1[(k >> 2) & 1]
                    * t2[(k >> 1) & 1] * t3[k & 1];
            R[k * 16 + lane] = r;
        }
    }
    __syncthreads();

    // ---- WMMA operands -----------------------------------------------------
    // A (16x4 f32): lanes 0-15 hold K=0,1 of row M=lane; lanes 16-31 K=2,3.
    // B (4x16 f32): VGPR0 = row (0|2) by lane half, VGPR1 = row (1|3).
    int nb   = lane & 15;
    int koff = (lane >> 4) * 2;

    v2f bmat[4];
#pragma unroll
    for (int j = 0; j < 4; ++j) {
        int kb = 4 * j + koff;
        bmat[j][0] = R[kb * 16 + nb];
        bmat[j][1] = R[(kb + 1) * 16 + nb];
    }

    const float* Pb = PQ + c * 512;
    const float* Qb = Pb + 256;
    int arow = lane & 15;

    v8f u = {0.f, 0.f, 0.f, 0.f, 0.f, 0.f, 0.f, 0.f};
    v8f v = {0.f, 0.f, 0.f, 0.f, 0.f, 0.f, 0.f, 0.f};
#pragma unroll
    for (int j = 0; j < 4; ++j) {
        v2f a;
        a[0] = Pb[arow * 16 + 4 * j + koff];
        a[1] = Pb[arow * 16 + 4 * j + koff + 1];
        u = __builtin_amdgcn_wmma_f32_16x16x4_f32(false, a, false, bmat[j],
                                                  (short)0, u, false, false);
    }
#pragma unroll
    for (int j = 0; j < 4; ++j) {
        v2f a;
        a[0] = Qb[arow * 16 + 4 * j + koff];
        a[1] = Qb[arow * 16 + 4 * j + koff + 1];
        v = __builtin_amdgcn_wmma_f32_16x16x4_f32(false, a, false, bmat[j],
                                                  (short)0, v, false, false);
    }

    // ---- probabilities + Z expectation sums --------------------------------
    // D layout: VGPR d = row M = d + (lane>=16 ? 8 : 0), col N = lane&15.
    int mhi = (lane >> 4) * 8;
    float sq0 = 0.f, sq1 = 0.f, sq2 = 0.f, sq3 = 0.f;
#pragma unroll
    for (int d = 0; d < 8; ++d) {
        int m = mhi + d;
        float p = u[d] * u[d] + v[d] * v[d];
        sq0 += ((m >> 3) & 1) ? -p : p;
        sq1 += ((m >> 2) & 1) ? -p : p;
        sq2 += ((m >> 1) & 1) ? -p : p;
        sq3 += (m & 1)        ? -p : p;
    }
    // fold rows 0-7 (lanes 0-15) with rows 8-15 (lanes 16-31)
    sq0 += __shfl_xor(sq0, 16, 32);
    sq1 += __shfl_xor(sq1, 16, 32);
    sq2 += __shfl_xor(sq2, 16, 32);
    sq3 += __shfl_xor(sq3, 16, 32);

    if (lane < 16) {
        int b = b0 + lane;
        float4 o = make_float4(sq0, sq1, sq2, sq3);
        *(float4*)(out + (size_t)b * 1024 + c * 4) = o;
    }
}

// ---------------------------------------------------------------------------
extern "C" void kernel_launch(void* const* d_in, const int* in_sizes, int n_in,
                              void* d_out, int out_size, void* d_ws, size_t ws_size,
                              hipStream_t stream) {
    const float* x = (const float*)d_in[0];       // (4096, 1024)
    const float* w = (const float*)d_in[1];       // (256, 2, 4)
    float* out = (float*)d_out;                   // (4096, 1024)
    float* PQ  = (float*)d_ws;                    // 256 * 512 floats = 512 KB

    build_pq_kernel<<<16, 256, 0, stream>>>(w, PQ);
    quantum_expval_kernel<<<(NQB / 16) * NCIR / 8, 256, 0, stream>>>(x, PQ, out);
}